// TopKPooling_76733885710649
// MI455X (gfx1250) — compile-verified
//
#include <hip/hip_runtime.h>
#include <math.h>

// ---------------- static workload config (matches reference) ----------------
constexpr int   B_G      = 64;
constexpr int   N_PER    = 8192;
constexpr int   NUM_SENT = 128;
constexpr int   C_DIM    = 256;
constexpr long  E_CNT    = 4194304;
constexpr int   F_EDGE   = 8;
constexpr int   K_SENT   = 3;
constexpr int   K_OTHER  = 4032;                       // ceil(0.5 * 8064)
constexpr int   P_TOT    = B_G * (K_SENT + K_OTHER);   // 258240
constexpr long  NTOT     = (long)B_G * N_PER;          // 524288
constexpr int   NBLK_E   = 1024;                       // edge blocks (4096 edges each)
constexpr int   EPB      = 4096;

// d_out layout (floats, concatenated in reference return order)
constexpr long OFF_X     = 0;
constexpr long OFF_EI    = (long)P_TOT * C_DIM;            // 2*E ints-as-float
constexpr long OFF_EA    = OFF_EI + 2 * E_CNT;             // E*8 floats
constexpr long OFF_BATCH = OFF_EA + E_CNT * F_EDGE;        // P
constexpr long OFF_PERM  = OFF_BATCH + P_TOT;              // P
constexpr long OFF_SCORE = OFF_PERM + P_TOT;               // P

typedef __attribute__((ext_vector_type(2))) float v2f;
typedef __attribute__((ext_vector_type(8))) float v8f;

__device__ __forceinline__ unsigned orderable(float f) {
    unsigned u = __float_as_uint(f);
    return (u & 0x80000000u) ? ~u : (u | 0x80000000u);   // monotonic ascending
}

// ---------------- kernel 0: 1/||w|| ----------------
__global__ void knorm(const float* __restrict__ w, float* __restrict__ invn) {
    __shared__ float red[256];
    int tid = threadIdx.x;
    float v = w[tid];
    red[tid] = v * v;
    __syncthreads();
    for (int s = 128; s > 0; s >>= 1) {
        if (tid < s) red[tid] += red[tid + s];
        __syncthreads();
    }
    if (tid == 0) invn[0] = rsqrtf(red[0]);
}

// ---------------- kernel 1: scores via V_WMMA_F32_16X16X4_F32 ----------------
// Each wave computes 16 rows' dot(X_row, w) by accumulating 16x16x4 WMMAs with
// w broadcast into column 0 of B. Block = 8 waves = 128 rows.
// B fragment is built branchlessly: unconditional ds_load of the w pair times a
// per-lane 0/1 mask (avoids divergent exec-predicated global loads in the loop).
__global__ __launch_bounds__(256) void kscore(const float* __restrict__ x,
                                              const float* __restrict__ w,
                                              const float* __restrict__ invnp,
                                              float* __restrict__ scores) {
    __shared__ float wl[C_DIM];
    wl[threadIdx.x] = w[threadIdx.x];
    __syncthreads();

    const int lane = threadIdx.x & 31;
    const int wv   = threadIdx.x >> 5;
    const int m    = lane & 15;          // A row within tile / B column
    const int kk   = (lane >> 4) * 2;    // K sub-offset {0,2}
    const long R   = (long)blockIdx.x * 128 + wv * 16;
    const float colMask = (m == 0) ? 1.0f : 0.0f;   // only column 0 of B is live

    v8f c = {};
    const float* arow = x + (R + m) * C_DIM + kk;
    for (int k0 = 0; k0 < C_DIM; k0 += 4) {
        v2f a, b;
        a[0] = arow[k0];
        a[1] = arow[k0 + 1];
        b[0] = wl[k0 + kk]     * colMask;
        b[1] = wl[k0 + kk + 1] * colMask;
        c = __builtin_amdgcn_wmma_f32_16x16x4_f32(false, a, false, b,
                                                  (short)0, c, false, false);
    }
    // D[M][0] lives in lane 0 (M = v, v=0..7) and lane 16 (M = v+8)
    const float invn = invnp[0];
    if (m == 0) {
        const int half = lane >> 4;
        for (int v = 0; v < 8; ++v)
            scores[R + half * 8 + v] = tanhf(c[v] * invn);
    }
}

// ---------------- kernel 2: per-graph bitonic top-k (one block per graph) ---
// key = (~orderable(score) << 32) | local_index  -> ascending sort gives
// descending score, ascending index on ties (matches stable argsort(-score)).
__global__ __launch_bounds__(1024) void ksort(const float* __restrict__ scores,
                                              float* __restrict__ out,
                                              int* __restrict__ perm_i) {
    __shared__ unsigned long long key[N_PER];   // 64 KB (WGP has 320 KB LDS)
    const int b   = blockIdx.x;
    const int tid = threadIdx.x;
    const long sb = (long)b * N_PER;

    // ---- sentence pool: sort 128 ----
    if (tid < NUM_SENT) {
        float s = scores[sb + tid];
        key[tid] = ((unsigned long long)(~orderable(s)) << 32) | (unsigned)tid;
    }
    __syncthreads();
    for (int k = 2; k <= NUM_SENT; k <<= 1)
        for (int j = k >> 1; j > 0; j >>= 1) {
            if (tid < NUM_SENT) {
                int ixj = tid ^ j;
                if (ixj > tid) {
                    bool up = ((tid & k) == 0);
                    unsigned long long a = key[tid], c = key[ixj];
                    if ((a > c) == up) { key[tid] = c; key[ixj] = a; }
                }
            }
            __syncthreads();
        }
    if (tid < K_SENT) {
        int local = (int)(key[tid] & 0xFFFFFFFFu);
        int g     = (int)sb + local;
        int pos   = b * K_SENT + tid;
        out[OFF_PERM  + pos] = (float)g;
        out[OFF_SCORE + pos] = scores[sb + local];
        out[OFF_BATCH + pos] = (float)b;
        perm_i[pos] = g;
    }
    __syncthreads();

    // ---- other pool: sort 8064 padded to 8192 ----
    for (int j = tid; j < N_PER; j += 1024) {
        if (j < N_PER - NUM_SENT) {
            float s = scores[sb + NUM_SENT + j];
            key[j] = ((unsigned long long)(~orderable(s)) << 32) | (unsigned)j;
        } else {
            key[j] = ~0ull;   // pads sort to the end
        }
    }
    __syncthreads();
    for (int k = 2; k <= N_PER; k <<= 1)
        for (int j = k >> 1; j > 0; j >>= 1) {
            for (int i = tid; i < N_PER; i += 1024) {
                int ixj = i ^ j;
                if (ixj > i) {
                    bool up = ((i & k) == 0);
                    unsigned long long a = key[i], c = key[ixj];
                    if ((a > c) == up) { key[i] = c; key[ixj] = a; }
                }
            }
            __syncthreads();
        }
    for (int j = tid; j < K_OTHER; j += 1024) {
        int local = (int)(key[j] & 0xFFFFFFFFu);
        int nl    = NUM_SENT + local;
        int g     = (int)sb + nl;
        int pos   = B_G * K_SENT + b * K_OTHER + j;
        out[OFF_PERM  + pos] = (float)g;
        out[OFF_SCORE + pos] = scores[sb + nl];
        out[OFF_BATCH + pos] = (float)b;
        perm_i[pos] = g;
    }
}

// ---------------- kernel 3/4: node_map build ----------------
__global__ void kinit(int* __restrict__ nmap) {
    long i = (long)blockIdx.x * 256 + threadIdx.x;
    if (i < NTOT) nmap[i] = -1;
}
__global__ void kscatter(const int* __restrict__ perm_i, int* __restrict__ nmap) {
    int p = blockIdx.x * 256 + threadIdx.x;
    if (p < P_TOT) nmap[perm_i[p]] = p;
}

// ---------------- kernel 5: x_out = x[perm] * score (wave per row) ---------
__global__ __launch_bounds__(256) void kgather(const float* __restrict__ x,
                                               const int* __restrict__ perm_i,
                                               const float* __restrict__ scoreSeg,
                                               float* __restrict__ xout) {
    const int wv = threadIdx.x >> 5, lane = threadIdx.x & 31;
    const long p = (long)blockIdx.x * 8 + wv;
    const int  node = perm_i[p];
    const float s   = scoreSeg[p];
    const float4* __restrict__ src = (const float4*)(x + (long)node * C_DIM);
    float4* __restrict__ dst       = (float4*)(xout + p * C_DIM);
    __builtin_prefetch(src + lane, 0, 1);   // global_prefetch_b8
    for (int t = 0; t < 2; ++t) {
        int idx = t * 32 + lane;            // 64 float4 per 256-float row
        float4 v = src[idx];
        v.x *= s; v.y *= s; v.z *= s; v.w *= s;
        dst[idx] = v;
    }
}

// ---------------- kernel 6: per-block valid-edge counts ----------------
__global__ __launch_bounds__(256) void kecount(const int* __restrict__ ei,
                                               const int* __restrict__ nmap,
                                               int* __restrict__ boffs) {
    __shared__ int red[256];
    const int tid = threadIdx.x;
    const long base = (long)blockIdx.x * EPB;
    int cnt = 0;
    for (int t = 0; t < 16; ++t) {
        long gid = base + t * 256 + tid;
        int r = nmap[ei[gid]];
        int c = nmap[ei[E_CNT + gid]];
        cnt += (r >= 0 && c >= 0) ? 1 : 0;
    }
    red[tid] = cnt;
    __syncthreads();
    for (int s = 128; s > 0; s >>= 1) {
        if (tid < s) red[tid] += red[tid + s];
        __syncthreads();
    }
    if (tid == 0) boffs[blockIdx.x] = red[0];
}

// ---------------- kernel 7: exclusive scan of 1024 counts ----------------
__global__ __launch_bounds__(1024) void kescan(int* __restrict__ boffs) {
    __shared__ int s[NBLK_E];
    const int tid = threadIdx.x;
    int v = boffs[tid];
    s[tid] = v;
    __syncthreads();
    for (int d = 1; d < NBLK_E; d <<= 1) {
        int t = (tid >= d) ? s[tid - d] : 0;
        __syncthreads();
        s[tid] += t;
        __syncthreads();
    }
    boffs[tid] = s[tid] - v;                 // exclusive prefix
    if (tid == NBLK_E - 1) boffs[NBLK_E] = s[NBLK_E - 1];   // total valid V
}

// ---------------- kernel 8: stable partition scatter of edges ----------------
__global__ __launch_bounds__(256) void kescatter(const int* __restrict__ ei,
                                                 const float* __restrict__ ea,
                                                 const int* __restrict__ nmap,
                                                 const int* __restrict__ boffs,
                                                 float* __restrict__ out) {
    __shared__ int waveTot[8];
    const int tid = threadIdx.x, lane = tid & 31, wv = tid >> 5;
    const long base = (long)blockIdx.x * EPB;
    long runValid = boffs[blockIdx.x];
    const long V  = boffs[NBLK_E];

    for (int t = 0; t < 16; ++t) {
        long gid = base + t * 256 + tid;
        int r = nmap[ei[gid]];
        int c = nmap[ei[E_CNT + gid]];
        bool valid = (r >= 0) && (c >= 0);

        unsigned mask = (unsigned)__ballot(valid);          // wave32
        int lanePref  = __popc(mask & ((1u << lane) - 1u));
        if (lane == 0) waveTot[wv] = __popc(mask);
        __syncthreads();
        int wOff = 0, chunkTot = 0;
        for (int i = 0; i < 8; ++i) {
            int w = waveTot[i];
            if (i < wv) wOff += w;
            chunkTot += w;
        }
        __syncthreads();   // waveTot reused next chunk

        long validPrefix = runValid + wOff + lanePref;      // # valid strictly before gid
        long pos = valid ? validPrefix : V + (gid - validPrefix);

        out[OFF_EI + pos]         = valid ? (float)r : -1.0f;
        out[OFF_EI + E_CNT + pos] = valid ? (float)c : -1.0f;

        const float4* __restrict__ eap = (const float4*)(ea + gid * F_EDGE);
        float4 z = {0.f, 0.f, 0.f, 0.f};
        float4 a0 = valid ? eap[0] : z;
        float4 a1 = valid ? eap[1] : z;
        float4* __restrict__ op = (float4*)(out + OFF_EA + pos * F_EDGE);
        op[0] = a0;
        op[1] = a1;

        runValid += chunkTot;
    }
}

// ---------------- host launcher ----------------
extern "C" void kernel_launch(void* const* d_in, const int* in_sizes, int n_in,
                              void* d_out, int out_size, void* d_ws, size_t ws_size,
                              hipStream_t stream) {
    const float* x  = (const float*)d_in[0];
    const float* w  = (const float*)d_in[1];
    const int*   ei = (const int*)d_in[2];
    const float* ea = (const float*)d_in[3];
    float* out = (float*)d_out;

    // workspace layout (4-byte units): [invn(16)] [scores NTOT] [nmap NTOT] [perm P] [boffs NBLK+1]
    float* wsf    = (float*)d_ws;
    float* invn   = wsf;
    float* scores = wsf + 16;
    int*   nmap   = (int*)(wsf + 16 + NTOT);
    int*   perm_i = nmap + NTOT;
    int*   boffs  = perm_i + P_TOT;

    knorm   <<<1, 256, 0, stream>>>(w, invn);
    kscore  <<<(int)(NTOT / 128), 256, 0, stream>>>(x, w, invn, scores);
    ksort   <<<B_G, 1024, 0, stream>>>(scores, out, perm_i);
    kinit   <<<(int)(NTOT / 256), 256, 0, stream>>>(nmap);
    kscatter<<<(P_TOT + 255) / 256, 256, 0, stream>>>(perm_i, nmap);
    kgather <<<P_TOT / 8, 256, 0, stream>>>(x, perm_i, out + OFF_SCORE, out + OFF_X);
    kecount <<<NBLK_E, 256, 0, stream>>>(ei, nmap, boffs);
    kescan  <<<1, NBLK_E, 0, stream>>>(boffs);
    kescatter<<<NBLK_E, 256, 0, stream>>>(ei, ea, nmap, boffs, out);
}